// GAT_Processor_10239202033755
// MI455X (gfx1250) — compile-verified
//
#include <hip/hip_runtime.h>
#include <math.h>

typedef __attribute__((ext_vector_type(2))) float v2f;
typedef __attribute__((ext_vector_type(8))) float v8f;

#define D 64
#define NEG 0.2f

// ---------- helpers: order-preserving float<->uint key for atomicMax ----------
__device__ __forceinline__ unsigned fkey(float f) {
    unsigned u = __float_as_uint(f);
    return (u & 0x80000000u) ? ~u : (u | 0x80000000u);
}
__device__ __forceinline__ float funkey(unsigned k) {
    return (k & 0x80000000u) ? __uint_as_float(k & 0x7fffffffu) : __uint_as_float(~k);
}

// ---------- ve[l][k] = sum_d We[l][k][d] * a_e[l][d]  (tiny) ----------
__global__ void k_ve(const float* __restrict__ We, const float* __restrict__ ae,
                     float* __restrict__ ve) {
    int l = blockIdx.x, k = threadIdx.x;
    const float* w = We + (size_t)l * D * D + (size_t)k * D;
    const float* a = ae + (size_t)l * D;
    float s = 0.f;
    for (int d = 0; d < D; ++d) s += w[d] * a[d];
    ve[l * D + k] = s;
}

// ---------- per-edge dots with ve (3 layers at once) + self-loop mean accum ----------
// wave-per-edge: lanes load float2 (coalesced 256B), shuffle-reduce.
__global__ void k_edot(const float* __restrict__ eattr, const int* __restrict__ ei,
                       const float* __restrict__ ve, float* __restrict__ edot,
                       float* __restrict__ selfsum, float* __restrict__ cnt,
                       int E, int N) {
    int wv = (blockIdx.x * blockDim.x + threadIdx.x) >> 5;
    int lane = threadIdx.x & 31;
    if (wv >= E) return;
    v2f e = *(const v2f*)(eattr + (size_t)wv * D + 2 * lane);
    float p[3];
    #pragma unroll
    for (int l = 0; l < 3; ++l) {
        v2f v = *(const v2f*)(ve + l * D + 2 * lane);
        p[l] = e.x * v.x + e.y * v.y;
    }
    #pragma unroll
    for (int off = 16; off; off >>= 1) {
        #pragma unroll
        for (int l = 0; l < 3; ++l) p[l] += __shfl_down(p[l], off, 32);
    }
    if (lane == 0) {
        int dst = ei[(size_t)E + wv];
        #pragma unroll
        for (int l = 0; l < 3; ++l) {
            edot[(size_t)l * E + wv] = p[l];
            atomicAdd(&selfsum[(size_t)l * N + dst], p[l]);
        }
        atomicAdd(&cnt[dst], 1.0f);
    }
}

// ---------- selfsum -> selfdot in place: /= max(cnt,1) ----------
__global__ void k_selfdot(float* __restrict__ selfsum, const float* __restrict__ cnt, int N) {
    size_t t = (size_t)blockIdx.x * blockDim.x + threadIdx.x;
    if (t >= (size_t)3 * N) return;
    int n = (int)(t % N);
    float c = cnt[n];
    selfsum[t] /= (c > 1.0f ? c : 1.0f);
}

// ---------- h = (relu?)x @ W via V_WMMA_F32_16X16X4_F32 ----------
// One wave computes one 16x16 tile of h. N % 16 == 0 (100000 = 6250*16).
// A 16x4 f32 frag: lanes 0-15 hold M=lane, K={k0,k0+1}; lanes 16-31 K={k0+2,k0+3}.
// B 4x16 f32 frag: lanes 0-15 hold N=lane, K={k0,k0+1}; lanes 16-31 K={k0+2,k0+3}.
// C/D f32: VGPR r, lanes 0-15 -> (M=r, N=lane); lanes 16-31 -> (M=r+8, N=lane-16).
__global__ void k_gemm(const float* __restrict__ x, const float* __restrict__ W,
                       float* __restrict__ h, int relu_in, int ntm) {
    int wv = (blockIdx.x * blockDim.x + threadIdx.x) >> 5;
    int lane = threadIdx.x & 31;
    if (wv >= ntm * 4) return;                 // uniform per wave: EXEC all-ones inside
    int mt = wv >> 2, nt = wv & 3;
    int m0 = mt * 16, n0 = nt * 16;
    int hl = lane >> 4;                        // half-wave id
    int lid = lane & 15;
    v8f acc = {};
    #pragma unroll
    for (int k0 = 0; k0 < D; k0 += 4) {
        const float* ap = x + (size_t)(m0 + lid) * D + k0 + hl * 2;
        v2f a; a.x = ap[0]; a.y = ap[1];
        if (relu_in) { a.x = a.x > 0.f ? a.x : 0.f; a.y = a.y > 0.f ? a.y : 0.f; }
        const float* bp = W + (size_t)(k0 + hl * 2) * D + n0 + lid;
        v2f b; b.x = bp[0]; b.y = bp[D];
        acc = __builtin_amdgcn_wmma_f32_16x16x4_f32(false, a, false, b,
                                                    (short)0, acc, false, false);
    }
    float* hp = h + (size_t)(m0 + hl * 8) * D + n0 + lid;
    #pragma unroll
    for (int r = 0; r < 8; ++r) hp[(size_t)r * D] = acc[r];
}

// ---------- s[n]=h[n]·a_s, d[n]=h[n]·a_d; init amax/denom ----------
__global__ void k_sd(const float* __restrict__ h, const float* __restrict__ as_,
                     const float* __restrict__ ad_, float* __restrict__ s,
                     float* __restrict__ d, unsigned* __restrict__ amax,
                     float* __restrict__ denom, int N) {
    int wv = (blockIdx.x * blockDim.x + threadIdx.x) >> 5;
    int lane = threadIdx.x & 31;
    if (wv >= N) return;
    v2f hv = *(const v2f*)(h + (size_t)wv * D + 2 * lane);
    v2f av = *(const v2f*)(as_ + 2 * lane);
    v2f dv = *(const v2f*)(ad_ + 2 * lane);
    float ps = hv.x * av.x + hv.y * av.y;
    float pd = hv.x * dv.x + hv.y * dv.y;
    #pragma unroll
    for (int off = 16; off; off >>= 1) {
        ps += __shfl_down(ps, off, 32);
        pd += __shfl_down(pd, off, 32);
    }
    if (lane == 0) { s[wv] = ps; d[wv] = pd; amax[wv] = 0u; denom[wv] = 0.f; }
}

// ---------- out[n][d] = bias[d] ----------
__global__ void k_init_out(float* __restrict__ out, const float* __restrict__ b, int N) {
    size_t t = (size_t)blockIdx.x * blockDim.x + threadIdx.x;
    if (t >= (size_t)N * D) return;
    out[t] = b[t & (D - 1)];
}

// ---------- alpha = leaky(s[src]+d[dst]+edot); atomicMax per dst ----------
__global__ void k_alpha(const int* __restrict__ ei, const float* __restrict__ edot,
                        const float* __restrict__ selfdot, const float* __restrict__ s,
                        const float* __restrict__ d, float* __restrict__ alpha,
                        unsigned* __restrict__ amax, int E, int N) {
    int j = blockIdx.x * blockDim.x + threadIdx.x;
    if (j >= E + N) return;
    int src, dst; float ed;
    if (j < E) { src = ei[j]; dst = ei[(size_t)E + j]; ed = edot[j]; }
    else       { src = dst = j - E;                    ed = selfdot[j - E]; }
    float a = s[src] + d[dst] + ed;
    a = a >= 0.f ? a : NEG * a;
    alpha[j] = a;
    atomicMax(&amax[dst], fkey(a));
}

// ---------- ea = exp(alpha - m[dst]); denom += ea ----------
__global__ void k_expsum(const int* __restrict__ ei, float* __restrict__ alpha,
                         const unsigned* __restrict__ amax, float* __restrict__ denom,
                         int E, int N) {
    int j = blockIdx.x * blockDim.x + threadIdx.x;
    if (j >= E + N) return;
    int dst = (j < E) ? ei[(size_t)E + j] : j - E;
    float e = __expf(alpha[j] - funkey(amax[dst]));
    alpha[j] = e;
    atomicAdd(&denom[dst], e);
}

// ---------- out[dst] += (ea/denom[dst]) * h[src]   (32 threads/edge, float2 each) ----------
__global__ void k_scatter(const int* __restrict__ ei, const float* __restrict__ ea,
                          const float* __restrict__ denom, const float* __restrict__ h,
                          float* __restrict__ out, int E, int N) {
    size_t t = (size_t)blockIdx.x * blockDim.x + threadIdx.x;
    int j = (int)(t >> 5);
    if (j >= E + N) return;
    int k = (int)(t & 31);
    int src, dst;
    if (j < E) { src = ei[j]; dst = ei[(size_t)E + j]; }
    else       { src = dst = j - E; }
    float coef = ea[j] / denom[dst];
    v2f hv = *(const v2f*)(h + (size_t)src * D + 2 * k);
    atomicAdd(&out[(size_t)dst * D + 2 * k],     coef * hv.x);
    atomicAdd(&out[(size_t)dst * D + 2 * k + 1], coef * hv.y);
}

static inline size_t align256(size_t x) { return (x + 255) & ~(size_t)255; }

extern "C" void kernel_launch(void* const* d_in, const int* in_sizes, int n_in,
                              void* d_out, int out_size, void* d_ws, size_t ws_size,
                              hipStream_t stream) {
    const float* node_hidden = (const float*)d_in[0];
    const float* edge_hidden = (const float*)d_in[1];
    const float* edge_attr   = (const float*)d_in[2];
    const int*   edge_index  = (const int*)d_in[3];
    const float* W      = (const float*)d_in[4];
    const float* W_edge = (const float*)d_in[5];
    const float* att_s  = (const float*)d_in[6];
    const float* att_d  = (const float*)d_in[7];
    const float* att_e  = (const float*)d_in[8];
    const float* bias   = (const float*)d_in[9];

    const int N = in_sizes[0] / D;          // 100000
    const int E = in_sizes[2] / D;          // 1600000
    const int Ep = E + N;                   // edges incl. self loops

    // workspace carve-out
    char* p = (char*)d_ws;
    float*    ve      = (float*)p;    p += align256((size_t)3 * D * 4);
    float*    cnt     = (float*)p;    p += align256((size_t)N * 4);
    float*    selfsum = (float*)p;    p += align256((size_t)3 * N * 4);   // becomes selfdot
    float*    edot    = (float*)p;    p += align256((size_t)3 * E * 4);
    float*    h       = (float*)p;    p += align256((size_t)N * D * 4);
    float*    sbuf    = (float*)p;    p += align256((size_t)N * 4);
    float*    dbuf    = (float*)p;    p += align256((size_t)N * 4);
    unsigned* amax    = (unsigned*)p; p += align256((size_t)N * 4);
    float*    denom   = (float*)p;    p += align256((size_t)N * 4);
    float*    alpha   = (float*)p;    p += align256((size_t)Ep * 4);
    float*    bufA    = (float*)p;    p += align256((size_t)N * D * 4);
    float*    bufB    = (float*)p;    p += align256((size_t)N * D * 4);

    float* xout = (float*)d_out;                    // final layer output (tuple elem 0)

    // tuple elem 1: edge_hidden passthrough
    hipMemcpyAsync((float*)d_out + (size_t)N * D, edge_hidden,
                   (size_t)E * D * sizeof(float), hipMemcpyDeviceToDevice, stream);

    // --- one-time precompute ---
    hipMemsetAsync(cnt, 0, (size_t)N * 4, stream);
    hipMemsetAsync(selfsum, 0, (size_t)3 * N * 4, stream);
    k_ve<<<3, D, 0, stream>>>(W_edge, att_e, ve);
    {
        int waves = E, blocks = (waves + 7) / 8;
        k_edot<<<blocks, 256, 0, stream>>>(edge_attr, edge_index, ve, edot,
                                           selfsum, cnt, E, N);
    }
    k_selfdot<<<(3 * N + 255) / 256, 256, 0, stream>>>(selfsum, cnt, N);
    float* selfdot = selfsum;

    const int ntm = N / 16;                         // 6250 tile-rows
    const float* xin = node_hidden;

    for (int l = 0; l < 3; ++l) {
        float* out = (l == 0) ? bufA : (l == 1) ? bufB : xout;
        // h = (relu?) xin @ W[l]   (WMMA f32)
        {
            int waves = ntm * 4, blocks = (waves + 7) / 8;
            k_gemm<<<blocks, 256, 0, stream>>>(xin, W + (size_t)l * D * D, h,
                                               l > 0 ? 1 : 0, ntm);
        }
        // per-node s,d; init amax/denom
        k_sd<<<(N + 7) / 8, 256, 0, stream>>>(h, att_s + (size_t)l * D,
                                              att_d + (size_t)l * D,
                                              sbuf, dbuf, amax, denom, N);
        // out = bias
        k_init_out<<<((size_t)N * D + 255) / 256, 256, 0, stream>>>(out,
                                              bias + (size_t)l * D, N);
        // alpha + segment max
        k_alpha<<<(Ep + 255) / 256, 256, 0, stream>>>(edge_index,
                                              edot + (size_t)l * E,
                                              selfdot + (size_t)l * N,
                                              sbuf, dbuf, alpha, amax, E, N);
        // exp + segment sum
        k_expsum<<<(Ep + 255) / 256, 256, 0, stream>>>(edge_index, alpha, amax,
                                                       denom, E, N);
        // weighted scatter-add
        {
            size_t threads = (size_t)Ep * 32;
            k_scatter<<<(threads + 255) / 256, 256, 0, stream>>>(edge_index, alpha,
                                              denom, h, out, E, N);
        }
        xin = out;
    }
}